// UMA_13950053777975
// MI455X (gfx1250) — compile-verified
//
#include <hip/hip_runtime.h>
#include <hip/hip_bf16.h>
#include <math.h>

// Problem constants (match reference)
#define B_    16
#define L_    2048
#define D_    512
#define MAXC_ 1025   // L/2 + 1

typedef __attribute__((ext_vector_type(2))) float v2f;
typedef __attribute__((ext_vector_type(8))) float v8f;

// ---------------------------------------------------------------------------
// Kernel 1: w[b,l] = sigmoid(X[b,l,:] . w_lin + bias) via V_WMMA_F32_16X16X4_F32
// One wave (wave32) per 16-row tile of t. A = X tile (16 rows x 4 K), B = w_lin
// chunk replicated across all 16 N columns -> every column of D equals y[0..15].
// 512/4 = 128 accumulating WMMAs per tile.
// A 16x4 f32 layout (ISA 7.12.2): lanes 0-15 hold M=lane, K={0,1} in v[0],v[1];
// lanes 16-31 hold M=lane-16, K={2,3}.  B 4x16 mirrors with N across lanes.
// C/D 16x16 f32: VGPR r, lanes 0-15 -> M=r, lanes 16-31 -> M=r+8.
// ---------------------------------------------------------------------------
__global__ void uma_w_kernel(const float* __restrict__ X,
                             const float* __restrict__ wl,
                             const float* __restrict__ bl,
                             float* __restrict__ wbuf) {
  const int lane = threadIdx.x & 31;
  const int wave = threadIdx.x >> 5;
  const int tile = blockIdx.x * 8 + wave;          // B_*L_/16 = 2048 tiles
  const int b    = tile / (L_ / 16);
  const int t0   = (tile % (L_ / 16)) * 16;
  const int mrow = lane & 15;
  const int koff = (lane >> 4) << 1;               // 0 (lanes 0-15) or 2 (lanes 16-31)
  const float* rowp = X + ((size_t)b * L_ + t0 + mrow) * D_;

  v8f c = {};
  for (int k0 = 0; k0 < D_; k0 += 4) {
    const float2 av = *(const float2*)(rowp + k0 + koff);   // 8B aligned
    const float2 bv = *(const float2*)(wl   + k0 + koff);
    v2f a;  a[0] = av.x;  a[1] = av.y;
    v2f bb; bb[0] = bv.x; bb[1] = bv.y;
    // 8 args: (neg_a, A, neg_b, B, c_mod, C, reuse_a, reuse_b)
    c = __builtin_amdgcn_wmma_f32_16x16x4_f32(false, a, false, bb,
                                              (short)0, c, false, false);
  }

  const float bias = bl[0];
  // Extract column (all columns equal): lanes 0-7 own rows 0-7 (c[lane]),
  // lanes 16-23 own rows 8-15 (c[lane-16]).
  int r = -1, m = 0;
  if (lane < 8)                       { r = lane;      m = lane; }
  else if (lane >= 16 && lane < 24)   { r = lane - 16; m = 8 + (lane - 16); }
  if (r >= 0) {
    float z;
    switch (r) {
      case 0: z = c[0]; break; case 1: z = c[1]; break;
      case 2: z = c[2]; break; case 3: z = c[3]; break;
      case 4: z = c[4]; break; case 5: z = c[5]; break;
      case 6: z = c[6]; break; default: z = c[7]; break;
    }
    z += bias;
    wbuf[b * L_ + t0 + m] = 1.0f / (1.0f + expf(-z));
  }
}

// ---------------------------------------------------------------------------
// Kernel 2: per batch, find strict local minima of w, front-pack boundary
// positions via block prefix-scan (equivalent to the reference's sort-compact),
// emit starts/ends/count and olens_out.
//   starts = [0, m_0, m_1, ...];  ends = [m_0+2, m_1+2, ..., L]
// ---------------------------------------------------------------------------
__global__ void uma_seg_kernel(const float* __restrict__ wbuf,
                               int* __restrict__ starts,
                               int* __restrict__ ends,
                               int* __restrict__ cnt,
                               const int* __restrict__ olens,
                               int* __restrict__ olens_out) {
  __shared__ float wsh[L_];
  __shared__ int   csum[256];
  const int b = blockIdx.x, tid = threadIdx.x;

  for (int i = tid; i < L_; i += 256) wsh[i] = wbuf[b * L_ + i];
  __syncthreads();

  int flags = 0, my = 0;
#pragma unroll
  for (int i = 0; i < 8; ++i) {
    const int pos = tid * 8 + i;
    const bool f = (pos > 0) && (pos < L_ - 1) &&
                   (wsh[pos] < wsh[pos - 1]) && (wsh[pos] < wsh[pos + 1]);
    flags |= (int)f << i;
    my += (int)f;
  }
  csum[tid] = my;
  __syncthreads();
  for (int off = 1; off < 256; off <<= 1) {     // Hillis-Steele inclusive scan
    const int v = (tid >= off) ? csum[tid - off] : 0;
    __syncthreads();
    csum[tid] += v;
    __syncthreads();
  }
  const int excl = csum[tid] - my;
  const int nmin = csum[255];

  int idx = excl;
#pragma unroll
  for (int i = 0; i < 8; ++i) {
    if ((flags >> i) & 1) {
      const int pos = tid * 8 + i;
      starts[b * MAXC_ + 1 + idx] = pos;        // starts[0] reserved for t=0
      ends[b * MAXC_ + idx]       = pos + 2;    // pos <= L-2 so pos+2 <= L
      ++idx;
    }
  }
  if (tid == 0) {
    starts[b * MAXC_ + 0]   = 0;
    ends[b * MAXC_ + nmin]  = L_;               // min(L-1+2, L)
    cnt[b]                  = nmin + 1;
    olens_out[b] = (int)((float)olens[b] / (float)olens[0] * (float)MAXC_);
  }
}

// ---------------------------------------------------------------------------
// Kernel 3: out[b,m,d] = sum_{t in [s,e)} x[b,t,d]*w[b,t] / max(sum w, 1e-6)
// One block per (m,b); 2 channels per thread (float2). X re-read hits the
// 192MB L2 (X is 64MB); prefetch next row. Invalid rows (m>=count) write 0.
// ---------------------------------------------------------------------------
__global__ void uma_pool_kernel(const float* __restrict__ X,
                                const float* __restrict__ wbuf,
                                const int* __restrict__ starts,
                                const int* __restrict__ ends,
                                const int* __restrict__ cnt,
                                float* __restrict__ out) {
  const int m = blockIdx.x, b = blockIdx.y, tid = threadIdx.x;
  const int d0 = tid * 2;

  int s = 0, e = 0;
  if (m < cnt[b]) { s = starts[b * MAXC_ + m]; e = ends[b * MAXC_ + m]; }

  float accx = 0.f, accy = 0.f, wsum = 0.f;
  for (int t = s; t < e; ++t) {
    const float wt = wbuf[b * L_ + t];
    const float2 x2 = *(const float2*)(X + ((size_t)b * L_ + t) * D_ + d0);
    if (t + 1 < e)
      __builtin_prefetch(X + ((size_t)b * L_ + t + 1) * D_ + d0, 0, 0);
    accx += x2.x * wt;
    accy += x2.y * wt;
    wsum += wt;
  }
  const float den = fmaxf(wsum, 1e-6f);
  float2 o; o.x = accx / den; o.y = accy / den;
  *(float2*)(out + ((size_t)b * MAXC_ + m) * D_ + d0) = o;
}

// ---------------------------------------------------------------------------
extern "C" void kernel_launch(void* const* d_in, const int* in_sizes, int n_in,
                              void* d_out, int out_size, void* d_ws, size_t ws_size,
                              hipStream_t stream) {
  const float* X    = (const float*)d_in[0];   // [B, L, D] f32
  const float* wlin = (const float*)d_in[1];   // [D] f32
  const float* blin = (const float*)d_in[2];   // [1] f32
  const int*   olen = (const int*)d_in[3];     // [B] i32

  float* out = (float*)d_out;                                  // [B, MAXC, D] f32
  int*   olens_out = (int*)(out + (size_t)B_ * MAXC_ * D_);    // [B] i32 (bit-cast tail)

  // Workspace layout (needs ~260 KB)
  float* wbuf   = (float*)d_ws;                                // B*L floats
  int*   starts = (int*)((char*)d_ws + (size_t)B_ * L_ * 4);   // B*MAXC ints
  int*   ends   = starts + B_ * MAXC_;                         // B*MAXC ints
  int*   cntb   = ends + B_ * MAXC_;                           // B ints

  // K1: 2048 wave-tiles, 8 waves per 256-thread block -> 256 blocks
  uma_w_kernel<<<dim3((B_ * L_ / 16) / 8), 256, 0, stream>>>(X, wlin, blin, wbuf);
  // K2: one block per batch
  uma_seg_kernel<<<dim3(B_), 256, 0, stream>>>(wbuf, starts, ends, cntb, olen, olens_out);
  // K3: one block per (segment slot, batch)
  uma_pool_kernel<<<dim3(MAXC_, B_), 256, 0, stream>>>(X, wbuf, starts, ends, cntb, out);
}